// IDTCWT_30262339568382
// MI455X (gfx1250) — compile-verified
//
#include <hip/hip_runtime.h>

// ============================================================================
// IDTCWT (2-level inverse dual-tree complex wavelet transform), MI455X/gfx1250.
//
// Math folding: sfb1d = upfirdn(x,f,2,1) + periodic wrap + roll(1-L/2).
// With f = numpy_filter[::-1] this collapses to:
//   out[2q+r] = sum_{e=-2..2}  lo[(q+e) mod Nh] * F[5+2e-r][0]
//                            + hi[(q+e) mod Nh] * F[5+2e-r][1]
// where F is the ORIGINAL (analysis-order) table. 10 MACs/output.
//
// Pipeline: K1 (Q-shift, level 2): lows+highs1 -> lo1 (workspace ~50MB,
// L2-resident on the 192MB L2).  K2 (first stage): lo1+highs0 -> out,
// sum of 4 branches / 2.  Each kernel: stage raw highs tile to LDS once
// (GLOBAL_LOAD_ASYNC_TO_LDS_B32 when available; butterfly on the fly so each
// highs element crosses HBM exactly once for all 4 branches), VALU W-pass
// stencil, WMMA H-pass: D(16x16) = A_filter(16x24) x B_interm(24x16), 6
// chained V_WMMA_F32_16X16X4_F32 per tile.  Roofline: ~350MB HBM traffic
// -> ~15us at 23.3 TB/s; fp32 end-to-end (memory-bound, no reason to quantize).
// ============================================================================

typedef __attribute__((ext_vector_type(2))) float v2f;
typedef __attribute__((ext_vector_type(8))) float v8f;

#if defined(__AMDGCN__) && \
    __has_builtin(__builtin_amdgcn_global_load_async_to_lds_b32) && \
    __has_builtin(__builtin_amdgcn_s_wait_asynccnt)
#define USE_ASYNC_LDS 1
#else
#define USE_ASYNC_LDS 0
#endif

typedef __attribute__((address_space(1))) int g_int;   // global (prints as __device__)
typedef __attribute__((address_space(3))) int l_int;   // LDS    (prints as __shared__)

// Copy one dword global -> LDS (async path: GLOBAL_LOAD_ASYNC_TO_LDS_B32,
// tracked on ASYNCcnt, no VGPR round trip).
__device__ __forceinline__ void stage_b32(const float* g, float* l) {
#if USE_ASYNC_LDS
  __builtin_amdgcn_global_load_async_to_lds_b32((g_int*)g, (l_int*)l, 0, 0);
#else
  *l = *g;
#endif
}
__device__ __forceinline__ void stage_fence() {
#if USE_ASYNC_LDS
  __builtin_amdgcn_s_wait_asynccnt(0);   // wait this wave's async copies
#endif
}

// Analysis-order filter tables [family][m][tap][col].
// family 0 = FSfarras first stage (final kernel), 1 = Kingsbury Q-shift.
__constant__ float c_filt[2][2][10][2] = {
  { // FSF: _F1, _F2 (original order)
    { {0.0f,0.0f},
      {-0.08838834764832f,-0.01122679215254f},
      { 0.08838834764832f, 0.01122679215254f},
      { 0.69587998903400f, 0.08838834764832f},
      { 0.69587998903400f, 0.08838834764832f},
      { 0.08838834764832f,-0.69587998903400f},
      {-0.08838834764832f, 0.69587998903400f},
      { 0.01122679215254f,-0.08838834764832f},
      { 0.01122679215254f,-0.08838834764832f},
      {0.0f,0.0f} },
    { { 0.01122679215254f,0.0f},
      { 0.01122679215254f,0.0f},
      {-0.08838834764832f,-0.08838834764832f},
      { 0.08838834764832f,-0.08838834764832f},
      { 0.69587998903400f, 0.69587998903400f},
      { 0.69587998903400f,-0.69587998903400f},
      { 0.08838834764832f, 0.08838834764832f},
      {-0.08838834764832f, 0.08838834764832f},
      {0.0f, 0.01122679215254f},
      {0.0f,-0.01122679215254f} }
  },
  { // SF: _Q1, _Q2 (original order)
    { { 0.03516384f,0.0f},
      {0.0f,0.0f},
      {-0.08832942f,-0.11430184f},
      { 0.23389032f,0.0f},
      { 0.76027237f, 0.58751830f},
      { 0.58751830f,-0.76027237f},
      {0.0f, 0.23389032f},
      {-0.11430184f, 0.08832942f},
      {0.0f,0.0f},
      {0.0f,-0.03516384f} },
    { {0.0f,-0.03516384f},
      {0.0f,0.0f},
      {-0.11430184f, 0.08832942f},
      {0.0f, 0.23389032f},
      { 0.58751830f,-0.76027237f},
      { 0.76027237f, 0.58751830f},
      { 0.23389032f,0.0f},
      {-0.08832942f,-0.11430184f},
      {0.0f,0.0f},
      { 0.03516384f,0.0f} }
  }
};

// A-matrix element for the H-pass WMMA: out row i (0..15), input row j (0..23).
// Rows 0..11 = loW[q0-2 .. q0+9] (filter col 0), 12..23 = hW (col 1).
template<int FAM>
__device__ __forceinline__ float amat(int mf, int i, int j) {
  int col = 0, jj = j;
  if (j >= 12) { col = 1; jj = j - 12; }
  int e = jj - 2 - (i >> 1);
  if (e < -2 || e > 2) return 0.0f;
  return c_filt[FAM][mf][5 + 2 * e - (i & 1)][col];
}

// One fused sfb2d over all 4 (m,n) branches.
//   FAM:    filter family; FINAL: accumulate branches (*0.5) into one output
//   HH,WH:  half-res input dims (compile-time -> immediate offsets, 32-bit math)
// Tile: 16 output rows (H) x 96 output floats (32 w x 3 c). Block = 192 = 6
// wave32s, wave wv owns output columns [16*wv, 16*wv+16).
template<int FAM, bool FINAL, int HH, int WH>
__global__ __launch_bounds__(192)
void sfb2d_fused(const float* __restrict__ loIn, const float* __restrict__ hIn,
                 float* __restrict__ outP, int B)
{
  constexpr int OH   = 2 * HH;
  constexpr int OX   = 6 * WH;           // output floats per row (2*WH w * 3 c)
  constexpr unsigned slab = (unsigned)HH * WH * 3;

  const int tid  = threadIdx.x;
  const int lane = tid & 31;
  const int wv   = tid >> 5;             // 0..5
  const int bz   = blockIdx.z;
  const int ht   = blockIdx.y;
  const int xt   = blockIdx.x;

  const int qh0 = 8 * ht;
  const int qw0 = 16 * xt;
  const int ph0 = 16 * ht;
  const int x0  = 96 * xt;

  __shared__ float s_h[12][12][60];      // [(j*2+t)*3+band][qh row][qw*3+c]
  __shared__ float s_lo[12][60];
  __shared__ float s_loW[12][104];       // stride 104: conflict-free B reads
  __shared__ float s_hW[12][104];

  // ---- stage raw highs once (each HBM byte read exactly once for 4 branches)
  for (int k = 0; k < 45; ++k) {         // 45*192 == 12*12*60
    int u   = tid + k * 192;
    int a   = u / 720;
    int rem = u - a * 720;
    int row = rem / 60;
    int cc  = rem - row * 60;
    int qwl = cc / 3, c = cc - qwl * 3;
    int qh = qh0 - 2 + row; qh += (qh < 0) ? HH : 0; qh -= (qh >= HH) ? HH : 0;
    int qw = qw0 - 2 + qwl; qw += (qw < 0) ? WH : 0; qw -= (qw >= WH) ? WH : 0;
    unsigned go = (unsigned)(a * B + bz) * slab + (unsigned)(qh * WH + qw) * 3u + (unsigned)c;
    stage_b32(hIn + go, &s_h[a][row][cc]);
  }
  stage_fence();

  const int i   = lane & 15;             // WMMA: A row M == B/D col N
  const int hl  = lane >> 4;             // half-wave select
  const int col = wv * 16 + i;           // local output column (x)
  const float is2 = 0.70710678118654752f;

  v8f acc = {0.f,0.f,0.f,0.f,0.f,0.f,0.f,0.f};

  for (int mn = 0; mn < 4; ++mn) {
    const int m  = mn >> 1, n = mn & 1;
    const int xo = m ^ n;                // butterfly pair select
    const float sB = (m == 0) ? is2 : -is2;

    __syncthreads();                     // protect s_lo/s_loW from prev readers

    // ---- stage lo tile for this branch
    const float* loBr = loIn + (unsigned)(mn * B + bz) * slab;
    for (int k = 0; k < 4; ++k) {
      int u = tid + k * 192;
      if (u < 720) {
        int row = u / 60;
        int cc  = u - row * 60;
        int qwl = cc / 3, c = cc - qwl * 3;
        int qh = qh0 - 2 + row; qh += (qh < 0) ? HH : 0; qh -= (qh >= HH) ? HH : 0;
        int qw = qw0 - 2 + qwl; qw += (qw < 0) ? WH : 0; qw -= (qw >= WH) ? WH : 0;
        stage_b32(loBr + (unsigned)(qh * WH + qw) * 3u + (unsigned)c, &s_lo[row][cc]);
      }
    }
    stage_fence();

    // W-filter (index n) into registers (uniform scalar loads)
    float fw0[10], fw1[10];
    #pragma unroll
    for (int t = 0; t < 10; ++t) {
      fw0[t] = c_filt[FAM][n][t][0];
      fw1[t] = c_filt[FAM][n][t][1];
    }
    __syncthreads();

    // ---- W-pass: 1152 parity pairs (both r share the same 5 taps)
    #pragma unroll
    for (int k = 0; k < 6; ++k) {
      int arr = (k >= 3);                // 0: loW (lo,LH)  1: hW (HL,HH)
      int rem = tid + (k - 3 * arr) * 192; // 0..575 = 12 rows x 16 qw x 3 c
      int row = rem / 48;
      int p2  = rem - row * 48;
      int qw_rel = p2 / 3, c = p2 - qw_rel * 3;
      float out0 = 0.f, out1 = 0.f;
      #pragma unroll
      for (int e = -2; e <= 2; ++e) {
        int tc = (qw_rel + e + 2) * 3 + c;
        float lov, hiv;
        if (arr == 0) {
          lov = s_lo[row][tc];
          hiv = s_h[xo * 3 + 0][row][tc] * is2 + s_h[6 + (1 ^ xo) * 3 + 0][row][tc] * sB;
        } else {
          lov = s_h[xo * 3 + 1][row][tc] * is2 + s_h[6 + (1 ^ xo) * 3 + 1][row][tc] * sB;
          hiv = s_h[xo * 3 + 2][row][tc] * is2 + s_h[6 + (1 ^ xo) * 3 + 2][row][tc] * sB;
        }
        out0 = fmaf(lov, fw0[5 + 2 * e], fmaf(hiv, fw1[5 + 2 * e], out0));
        out1 = fmaf(lov, fw0[4 + 2 * e], fmaf(hiv, fw1[4 + 2 * e], out1));
      }
      float* dst = arr ? &s_hW[row][0] : &s_loW[row][0];
      dst[6 * qw_rel + c]     = out0;    // w = 2*qw_rel   (r=0)
      dst[6 * qw_rel + 3 + c] = out1;    // w = 2*qw_rel+1 (r=1)
    }
    __syncthreads();

    // ---- H-pass: 6 chained V_WMMA_F32_16X16X4_F32, D = A_filt(16x24)*B(24x16)
    v8f cin;
    if (FINAL) cin = acc;
    else       cin = (v8f){0.f,0.f,0.f,0.f,0.f,0.f,0.f,0.f};
    #pragma unroll
    for (int s = 0; s < 6; ++s) {
      int kk = s * 4 + hl * 2;           // K indices kk, kk+1 for this lane
      v2f a;
      a.x = amat<FAM>(m, i, kk);
      a.y = amat<FAM>(m, i, kk + 1);
      const float* src = (kk < 12) ? &s_loW[kk][col] : &s_hW[kk - 12][col];
      v2f bv;
      bv.x = src[0];
      bv.y = src[104];                   // next LDS row (stride 104)
      cin = __builtin_amdgcn_wmma_f32_16x16x4_f32(
              false, a, false, bv, (short)0, cin, false, false);
    }

    if (FINAL) {
      acc = cin;
    } else {
      // Per-branch store: one base, compile-time row offsets (v*OX immediates)
      float* op = outP + (unsigned)(mn * B + bz) * (unsigned)(OH * OX)
                       + (unsigned)(ph0 + hl * 8) * (unsigned)OX
                       + (unsigned)(x0 + col);
      #pragma unroll
      for (int v = 0; v < 8; ++v) op[v * OX] = cin[v];
    }
  }

  if (FINAL) {
    float* op = outP + (unsigned)bz * (unsigned)(OH * OX)
                     + (unsigned)(ph0 + hl * 8) * (unsigned)OX
                     + (unsigned)(x0 + col);
    #pragma unroll
    for (int v = 0; v < 8; ++v) op[v * OX] = 0.5f * acc[v];
  }
}

extern "C" void kernel_launch(void* const* d_in, const int* in_sizes, int n_in,
                              void* d_out, int out_size, void* d_ws, size_t ws_size,
                              hipStream_t stream) {
  const float* highs0 = (const float*)d_in[0];   // (2,2,3,B,256,256,3)
  const float* highs1 = (const float*)d_in[1];   // (2,2,3,B,128,128,3)
  const float* lows   = (const float*)d_in[2];   // (2,2,B,128,128,3)
  float* out = (float*)d_out;                    // (B,512,512,3)
  float* lo1 = (float*)d_ws;                     // workspace (4,B,256,256,3) ~50MB

  const int B = in_sizes[2] / (2 * 2 * 128 * 128 * 3);
  dim3 blk(192, 1, 1);

  // Level 2 (Q-shift): lows + highs1 -> lo1. Per-branch out: 256 x 768 floats.
  dim3 g1(768 / 96, 256 / 16, B);
  sfb2d_fused<1, false, 128, 128><<<g1, blk, 0, stream>>>(lows, highs1, lo1, B);

  // Level 1 (first stage): lo1 + highs0 -> out, sum of 4 branches * 0.5.
  dim3 g2(1536 / 96, 512 / 16, B);
  sfb2d_fused<0, true, 256, 256><<<g2, blk, 0, stream>>>(lo1, highs0, out, B);
}